// GraphNeuralNetPyG_Advanced_28552942583895
// MI455X (gfx1250) — compile-verified
//
#include <hip/hip_runtime.h>

typedef __attribute__((ext_vector_type(2))) float v2f;
typedef __attribute__((ext_vector_type(8))) float v8f;

#define NN   50000
#define EE   800000
#define GGG  64
#define NDIM 128
#define HIDC 256

static __host__ __device__ inline int cdiv(long a, long b){ return (int)((a + b - 1)/b); }

__device__ __forceinline__ int fenc(float f){ int i = __float_as_int(f); return i < 0 ? (i ^ 0x7fffffff) : i; }
__device__ __forceinline__ float fdec(int i){ return __int_as_float(i < 0 ? (i ^ 0x7fffffff) : i); }

// ---------------- init ----------------
__global__ void k_fill_f32(float* p, float v, int n){
  for (int i = blockIdx.x*blockDim.x + threadIdx.x; i < n; i += gridDim.x*blockDim.x) p[i] = v;
}
__global__ void k_fill_i32(int* p, int v, int n){
  for (int i = blockIdx.x*blockDim.x + threadIdx.x; i < n; i += gridDim.x*blockDim.x) p[i] = v;
}

// ---------------- edge_attr mean ----------------
__global__ void k_edge_mean(const float* __restrict__ ea, float* mean2, int E){
  float s0 = 0.f, s1 = 0.f;
  for (int e = blockIdx.x*blockDim.x + threadIdx.x; e < E; e += gridDim.x*blockDim.x){
    s0 += ea[2*e]; s1 += ea[2*e+1];
  }
  atomicAdd(&mean2[0], s0);
  atomicAdd(&mean2[1], s1);
}

// ---------------- WMMA fp32 GEMM:  C[M,256] = A[M,K] @ B[K,256] (+bias) ----------------
// block = 256 threads (8 waves). blockIdx.x = 16-row strip. Wave w -> col tiles 2w, 2w+1.
// A strip (16 x K) staged in LDS, shared by all 8 waves.
__global__ void __launch_bounds__(256)
k_gemm_wmma(const float* __restrict__ A, const float* __restrict__ B,
            const float* __restrict__ bias, float* __restrict__ C, int K){
  __shared__ float lA[16 * 256];
  const int tid = threadIdx.x;
  const int m0  = blockIdx.x * 16;
  for (int i = tid; i < 16*K; i += 256){
    int r = i / K, c = i - r*K;
    lA[r*K + c] = A[(size_t)(m0 + r)*K + c];
  }
  __syncthreads();
  const int wave = tid >> 5;
  const int lane = tid & 31;
  const int hi = lane >> 4;     // half-wave select
  const int lo = lane & 15;
  const int n0 = (wave*2)*16 + lo;
  const int n1 = n0 + 16;
  v8f acc0 = {}; v8f acc1 = {};
  for (int kk = 0; kk < K; kk += 4){
    const int kb = kk + 2*hi;
    v2f a; a.x = lA[lo*K + kb]; a.y = lA[lo*K + kb + 1];
    v2f b0, b1;
    b0.x = B[(size_t)kb*HIDC + n0];     b0.y = B[(size_t)(kb+1)*HIDC + n0];
    b1.x = B[(size_t)kb*HIDC + n1];     b1.y = B[(size_t)(kb+1)*HIDC + n1];
    acc0 = __builtin_amdgcn_wmma_f32_16x16x4_f32(false, a, false, b0, (short)0, acc0, false, false);
    acc1 = __builtin_amdgcn_wmma_f32_16x16x4_f32(false, a, false, b1, (short)0, acc1, false, false);
  }
  const float bv0 = bias ? bias[n0] : 0.f;
  const float bv1 = bias ? bias[n1] : 0.f;
  #pragma unroll
  for (int j = 0; j < 8; ++j){
    const int row = m0 + j + hi*8;
    C[(size_t)row*HIDC + n0] = acc0[j] + bv0;
    C[(size_t)row*HIDC + n1] = acc1[j] + bv1;
  }
}

// ---------------- per-node attention dots:  s[n,h] = sum_c h[n,h,c] * a[h,c] ----------------
__global__ void k_node_dots(const float* __restrict__ h, const float* __restrict__ a_s,
                            const float* __restrict__ a_d, float* __restrict__ ssrc,
                            float* __restrict__ sdst, int n, int H){
  int idx = blockIdx.x*blockDim.x + threadIdx.x;
  if (idx >= n*H) return;
  int node = idx / H, hd = idx - node*H;
  int C = HIDC / H;
  const float* hp = h + (size_t)node*HIDC + hd*C;
  const float* as = a_s + hd*C;
  const float* ad = a_d + hd*C;
  float s = 0.f, d = 0.f;
  for (int c = 0; c < C; ++c){ float v = hp[c]; s += v*as[c]; d += v*ad[c]; }
  ssrc[idx] = s; sdst[idx] = d;
}

// ---------------- edge-attr head constants:  A01[r*H+h] = sum_c We[r, h*C+c]*a_e[h,c] ----------------
__global__ void k_edge_consts(const float* __restrict__ We, const float* __restrict__ a_e,
                              float* __restrict__ A01, int H){
  int t = threadIdx.x;
  if (t >= 2*H) return;
  int r = t / H, hd = t - r*H;
  int C = HIDC / H;
  float s = 0.f;
  for (int c = 0; c < C; ++c) s += We[r*HIDC + hd*C + c] * a_e[hd*C + c];
  A01[r*H + hd] = s;
}

// ---------------- edge logits + leaky relu ----------------
__global__ void k_edge_logits(const int* __restrict__ ei, const float* __restrict__ ea,
                              const float* __restrict__ mean2, const float* __restrict__ A01,
                              const float* __restrict__ ssrc, const float* __restrict__ sdst,
                              float* __restrict__ al, int H){
  int idx = blockIdx.x*blockDim.x + threadIdx.x;
  int tot = (EE + NN) * H;
  if (idx >= tot) return;
  int e = idx / H, hd = idx - e*H;
  int sb, db; float e0, e1;
  if (e < EE){ sb = ei[e]; db = ei[EE + e]; e0 = ea[2*e]; e1 = ea[2*e+1]; }
  else { sb = db = e - EE; const float inv = 1.0f/(float)EE; e0 = mean2[0]*inv; e1 = mean2[1]*inv; }
  float v = ssrc[sb*H + hd] + sdst[db*H + hd] + e0*A01[hd] + e1*A01[H + hd];
  al[idx] = v > 0.f ? v : 0.2f*v;
}

// ---------------- segment softmax over destination ----------------
__global__ void k_seg_amax(const int* __restrict__ ei, const float* __restrict__ al,
                           int* __restrict__ amax, int H){
  int idx = blockIdx.x*blockDim.x + threadIdx.x;
  int tot = (EE + NN) * H;
  if (idx >= tot) return;
  int e = idx / H, hd = idx - e*H;
  int db = (e < EE) ? ei[EE + e] : (e - EE);
  atomicMax(&amax[db*H + hd], fenc(al[idx]));
}
__global__ void k_seg_expsum(const int* __restrict__ ei, float* __restrict__ al,
                             const int* __restrict__ amax, float* __restrict__ den, int H){
  int idx = blockIdx.x*blockDim.x + threadIdx.x;
  int tot = (EE + NN) * H;
  if (idx >= tot) return;
  int e = idx / H, hd = idx - e*H;
  int db = (e < EE) ? ei[EE + e] : (e - EE);
  float ex = __expf(al[idx] - fdec(amax[db*H + hd]));
  al[idx] = ex;
  atomicAdd(&den[db*H + hd], ex);
}
__global__ void k_alpha(const int* __restrict__ ei, float* __restrict__ al,
                        const float* __restrict__ den, int H){
  int idx = blockIdx.x*blockDim.x + threadIdx.x;
  int tot = (EE + NN) * H;
  if (idx >= tot) return;
  int e = idx / H, hd = idx - e*H;
  int db = (e < EE) ? ei[EE + e] : (e - EE);
  al[idx] = al[idx] / (den[db*H + hd] + 1e-16f);
}

// ---------------- aggregation: one wave32 per edge, 8 channels per lane ----------------
__global__ void __launch_bounds__(256)
k_aggregate(const int* __restrict__ ei, const float* __restrict__ h,
            const float* __restrict__ al, float* __restrict__ aggr, int H){
  int gid  = blockIdx.x*blockDim.x + threadIdx.x;
  int lane = threadIdx.x & 31;
  int e    = gid >> 5;
  if (e >= EE + NN) return;
  int sb, db;
  if (e < EE){ sb = ei[e]; db = ei[EE + e]; } else { sb = db = e - EE; }
  const int c0 = lane * 8;
  const int C  = HIDC / H;
  const int hd = c0 / C;
  const float alpha = al[(size_t)e*H + hd];
  const float4* hp = (const float4*)(h + (size_t)sb*HIDC + c0);
  float4 v0 = hp[0], v1 = hp[1];
  float* ap = aggr + (size_t)db*HIDC + c0;
  atomicAdd(ap+0, v0.x*alpha); atomicAdd(ap+1, v0.y*alpha);
  atomicAdd(ap+2, v0.z*alpha); atomicAdd(ap+3, v0.w*alpha);
  atomicAdd(ap+4, v1.x*alpha); atomicAdd(ap+5, v1.y*alpha);
  atomicAdd(ap+6, v1.z*alpha); atomicAdd(ap+7, v1.w*alpha);
}

// ---------------- bias + batchnorm(eval) + ELU + residual (in-place safe) ----------------
__global__ void k_bn_elu_res(const float* __restrict__ aggr, const float* __restrict__ bconv,
                             const float* __restrict__ g, const float* __restrict__ be,
                             const float* __restrict__ rm, const float* __restrict__ rv,
                             const float* __restrict__ identity, float* __restrict__ xout){
  int idx = blockIdx.x*blockDim.x + threadIdx.x;   // one thread = 4 channels
  if (idx >= NN * (HIDC/4)) return;
  int node = idx >> 6;
  int c0   = (idx & 63) * 4;
  #pragma unroll
  for (int k = 0; k < 4; ++k){
    int c = c0 + k;
    float v = aggr[(size_t)node*HIDC + c] + bconv[c];
    v = (v - rm[c]) * rsqrtf(rv[c] + 1e-5f) * g[c] + be[c];
    v = v > 0.f ? v : (__expf(v) - 1.f);
    xout[(size_t)node*HIDC + c] = v + identity[(size_t)node*HIDC + c];
  }
}

// ---------------- pooling ----------------
__global__ void k_count(const int* __restrict__ batch, float* __restrict__ cnt){
  int i = blockIdx.x*blockDim.x + threadIdx.x;
  if (i < NN) atomicAdd(&cnt[batch[i]], 1.f);
}
__global__ void k_pool(const float* __restrict__ x, const int* __restrict__ batch,
                       float* __restrict__ psum, int* __restrict__ pmax){
  int idx = blockIdx.x*blockDim.x + threadIdx.x;
  if (idx >= NN*HIDC) return;
  int node = idx >> 8, c = idx & 255;
  int b = batch[node];
  float v = x[idx];
  atomicAdd(&psum[b*HIDC + c], v);
  atomicMax(&pmax[b*HIDC + c], fenc(v));
}

// ---------------- tiny MLP head ----------------
__global__ void k_mlp1(const float* __restrict__ psum, const int* __restrict__ pmax,
                       const float* __restrict__ cnt, const float* __restrict__ Wp1,
                       const float* __restrict__ bp1, float* __restrict__ hmid){
  int g = blockIdx.x;       // 64
  int c = threadIdx.x;      // 256
  float cn = cnt[g]; cn = cn < 1.f ? 1.f : cn;
  const float inv = 1.f / cn;
  float acc = bp1[c];
  for (int k = 0; k < HIDC; ++k) acc += psum[g*HIDC + k] * inv * Wp1[k*HIDC + c];
  for (int k = 0; k < HIDC; ++k) acc += fdec(pmax[g*HIDC + k]) * Wp1[(HIDC + k)*HIDC + c];
  hmid[g*HIDC + c] = acc > 0.f ? acc : 0.f;
}
__global__ void k_mlp2(const float* __restrict__ hmid, const float* __restrict__ Wp2,
                       const float* __restrict__ bp2, float* __restrict__ out){
  int g = blockIdx.x*blockDim.x + threadIdx.x;
  if (g >= GGG) return;
  float acc = bp2[0];
  for (int c = 0; c < HIDC; ++c) acc += hmid[g*HIDC + c] * Wp2[c];
  out[g] = acc;
}

extern "C" void kernel_launch(void* const* d_in, const int* in_sizes, int n_in,
                              void* d_out, int out_size, void* d_ws, size_t ws_size,
                              hipStream_t stream){
  (void)in_sizes; (void)n_in; (void)out_size; (void)ws_size;
  const float* x   = (const float*)d_in[0];
  const int*   ei  = (const int*)  d_in[1];
  const float* ea  = (const float*)d_in[2];
  const int*   bat = (const int*)  d_in[3];
  const float *W[3], *asrc[3], *adst[3], *We[3], *aedge[3], *bconv[3],
              *bng[3], *bnb[3], *bnm[3], *bnv[3];
  for (int i = 0; i < 3; ++i){
    int b = 4 + i*10;
    W[i]     = (const float*)d_in[b+0];
    asrc[i]  = (const float*)d_in[b+1];
    adst[i]  = (const float*)d_in[b+2];
    We[i]    = (const float*)d_in[b+3];
    aedge[i] = (const float*)d_in[b+4];
    bconv[i] = (const float*)d_in[b+5];
    bng[i]   = (const float*)d_in[b+6];
    bnb[i]   = (const float*)d_in[b+7];
    bnm[i]   = (const float*)d_in[b+8];
    bnv[i]   = (const float*)d_in[b+9];
  }
  const float* Wskip = (const float*)d_in[34];
  const float* bskip = (const float*)d_in[35];
  const float* Wp1   = (const float*)d_in[36];
  const float* bp1   = (const float*)d_in[37];
  const float* Wp2   = (const float*)d_in[38];
  const float* bp2   = (const float*)d_in[39];
  float* out = (float*)d_out;

  // ---- workspace carve-up (all 16B aligned) ----
  const int    ET   = EE + NN;
  float* h     = (float*)d_ws;                  // NN*256
  float* xbuf  = h    + (size_t)NN*HIDC;        // NN*256
  float* aggr  = xbuf + (size_t)NN*HIDC;        // NN*256
  float* al    = aggr + (size_t)NN*HIDC;        // ET*4 max
  float* ssrc  = al   + (size_t)ET*4;           // NN*4
  float* sdst  = ssrc + (size_t)NN*4;           // NN*4
  int*   amax  = (int*)(sdst + (size_t)NN*4);   // NN*4
  float* den   = (float*)(amax + (size_t)NN*4); // NN*4
  float* mean2 = den  + (size_t)NN*4;           // 2
  float* A01   = mean2 + 4;                     // 8
  float* psum  = A01 + 12;                      // GGG*256
  int*   pmax  = (int*)(psum + (size_t)GGG*HIDC);
  float* cnt   = (float*)(pmax + (size_t)GGG*HIDC);
  float* hmid  = cnt + 64;                      // GGG*256

  const int NEG_INF_ENC = (int)0x807FFFFF;      // fenc(-inf)

  // edge_attr column means
  k_fill_f32<<<1, 64, 0, stream>>>(mean2, 0.f, 2);
  k_edge_mean<<<256, 256, 0, stream>>>(ea, mean2, EE);

  // skip connection:  xbuf = x @ Wskip + bskip   (fp32 WMMA)
  k_gemm_wmma<<<NN/16, 256, 0, stream>>>(x, Wskip, bskip, xbuf, NDIM);

  for (int i = 0; i < 3; ++i){
    const int H = (i < 2) ? 4 : 1;
    const int K = (i == 0) ? NDIM : HIDC;
    const float* in = (i == 0) ? x : xbuf;

    // h = in @ W[i]   (fp32 WMMA, LDS-staged A strip)
    k_gemm_wmma<<<NN/16, 256, 0, stream>>>(in, W[i], nullptr, h, K);

    // per-node / per-edge attention pieces
    k_node_dots<<<cdiv((long)NN*H, 256), 256, 0, stream>>>(h, asrc[i], adst[i], ssrc, sdst, NN, H);
    k_edge_consts<<<1, 64, 0, stream>>>(We[i], aedge[i], A01, H);
    k_edge_logits<<<cdiv((long)ET*H, 256), 256, 0, stream>>>(ei, ea, mean2, A01, ssrc, sdst, al, H);

    // segment softmax over dst
    k_fill_i32<<<cdiv((long)NN*H, 256), 256, 0, stream>>>(amax, NEG_INF_ENC, NN*H);
    k_fill_f32<<<cdiv((long)NN*H, 256), 256, 0, stream>>>(den, 0.f, NN*H);
    k_seg_amax  <<<cdiv((long)ET*H, 256), 256, 0, stream>>>(ei, al, amax, H);
    k_seg_expsum<<<cdiv((long)ET*H, 256), 256, 0, stream>>>(ei, al, amax, den, H);
    k_alpha     <<<cdiv((long)ET*H, 256), 256, 0, stream>>>(ei, al, den, H);

    // weighted scatter-add:  aggr[dst] += h[src] * alpha   (wave per edge)
    k_fill_f32<<<cdiv((long)NN*HIDC, 256), 256, 0, stream>>>(aggr, 0.f, NN*HIDC);
    k_aggregate<<<cdiv((long)ET*32, 256), 256, 0, stream>>>(ei, h, al, aggr, H);

    // bias + BN + ELU + residual  (identity = xbuf, in-place elementwise)
    k_bn_elu_res<<<cdiv((long)NN*(HIDC/4), 256), 256, 0, stream>>>(
        aggr, bconv[i], bng[i], bnb[i], bnm[i], bnv[i], xbuf, xbuf);
  }

  // pooling (mean + max per graph)
  k_fill_f32<<<cdiv((long)GGG*HIDC, 256), 256, 0, stream>>>(psum, 0.f, GGG*HIDC);
  k_fill_i32<<<cdiv((long)GGG*HIDC, 256), 256, 0, stream>>>(pmax, NEG_INF_ENC, GGG*HIDC);
  k_fill_f32<<<1, 64, 0, stream>>>(cnt, 0.f, GGG);
  k_count<<<cdiv(NN, 256), 256, 0, stream>>>(bat, cnt);
  k_pool <<<cdiv((long)NN*HIDC, 256), 256, 0, stream>>>(xbuf, bat, psum, pmax);

  // MLP head
  k_mlp1<<<GGG, HIDC, 0, stream>>>(psum, pmax, cnt, Wp1, bp1, hmid);
  k_mlp2<<<1, 64, 0, stream>>>(hmid, Wp2, bp2, out);
}